// GCN_encoder_54803782697391
// MI455X (gfx1250) — compile-verified
//
#include <hip/hip_runtime.h>

typedef float v2f __attribute__((ext_vector_type(2)));
typedef float v8f __attribute__((ext_vector_type(8)));

#define NHID 128            // hidden width (both layers' output width)
#define GEMM_WAVES 8        // waves per block -> 128 rows per block

#define WMMA_F32(ACC, AF, BF)                                              \
  ACC = __builtin_amdgcn_wmma_f32_16x16x4_f32(false, (AF), false, (BF),    \
                                              (short)0, ACC, false, false)

// ---------------------------------------------------------------------------
// g = (A[M,K] @ W[K,128]) * dinv[m]    using V_WMMA_F32_16X16X4_F32 (wave32)
// W is staged in LDS in a fragment-swizzled layout:
//   float2 slot [(kb*2 + c)*128 + n] = { W[4kb+2c+0, n], W[4kb+2c+1, n] }
// so lane L (c = L>=16, n = L%16 + 16*tile) reads its B fragment as one b64.
// Inner loop is software-pipelined one K-step ahead; an asm operand barrier
// keeps the current fragments register-resident so the 8 WMMAs issue with a
// single (partial) counter wait instead of a drain per pair.
// ---------------------------------------------------------------------------
__global__ __launch_bounds__(GEMM_WAVES * 32) void gemm_scale_kernel(
    const float* __restrict__ A, const float* __restrict__ W,
    const float* __restrict__ dinv, float* __restrict__ G, int M, int K) {
  extern __shared__ float lds[];  // K*128 floats

  const int tid = threadIdx.x;
  // cooperative load + swizzle of W into LDS
  for (int idx = tid; idx < K * NHID; idx += GEMM_WAVES * 32) {
    int k = idx >> 7;
    int n = idx & 127;
    int fidx = (((((k >> 2) << 1) + ((k >> 1) & 1)) << 7) + n) * 2 + (k & 1);
    lds[fidx] = W[idx];
  }
  __syncthreads();

  const int wave = tid >> 5;
  const int lane = tid & 31;
  const int tileM = blockIdx.x * GEMM_WAVES + wave;
  if (tileM * 16 >= M) return;  // wave-uniform exit (M%16==0 here -> full tiles)
  const int m0 = tileM * 16;
  const int lm = lane & 15;
  const int hi = lane >> 4;  // 0 or 1

  int arow_i = m0 + lm;
  if (arow_i > M - 1) arow_i = M - 1;  // clamp (generic tail safety)
  const float* arow = A + (size_t)arow_i * K + (hi << 1);

  v8f acc[8] = {};
  const int ksteps = K >> 2;

  // prologue: fragments for K-step 0
  const v2f* bp = ((const v2f*)lds) + hi * NHID + lm;
  v2f b0 = bp[0], b1 = bp[16], b2 = bp[32], b3 = bp[48];
  v2f b4 = bp[64], b5 = bp[80], b6 = bp[96], b7 = bp[112];
  v2f a = *(const v2f*)(arow);

  for (int kb = 0; kb < ksteps; ++kb) {
    // issue next K-step's loads first (clamped; harmless reload on last step)
    int nk = (kb + 1 < ksteps) ? kb + 1 : kb;
    const v2f* bn = ((const v2f*)lds) + ((nk << 1) + hi) * NHID + lm;
    v2f c0 = bn[0], c1 = bn[16], c2 = bn[32], c3 = bn[48];
    v2f c4 = bn[64], c5 = bn[80], c6 = bn[96], c7 = bn[112];
    v2f a2 = *(const v2f*)(arow + (nk << 2));

    // barrier: current fragments must be in registers; next-step loads
    // (issued above) cannot sink past this point.
    asm volatile("" : "+v"(b0), "+v"(b1), "+v"(b2), "+v"(b3), "+v"(b4),
                      "+v"(b5), "+v"(b6), "+v"(b7), "+v"(a)::"memory");

    WMMA_F32(acc[0], a, b0);
    WMMA_F32(acc[1], a, b1);
    WMMA_F32(acc[2], a, b2);
    WMMA_F32(acc[3], a, b3);
    WMMA_F32(acc[4], a, b4);
    WMMA_F32(acc[5], a, b5);
    WMMA_F32(acc[6], a, b6);
    WMMA_F32(acc[7], a, b7);

    b0 = c0; b1 = c1; b2 = c2; b3 = c3;
    b4 = c4; b5 = c5; b6 = c6; b7 = c7;
    a = a2;
  }

  // epilogue: scale rows by dinv[m], store
#pragma unroll
  for (int r = 0; r < 8; ++r) {
    int rr = m0 + r + (hi << 3);
    if (rr >= M) continue;
    float scl = dinv[rr];
    float* grow = G + (size_t)rr * NHID + lm;
#pragma unroll
    for (int t = 0; t < 8; ++t) grow[t * 16] = acc[t][r] * scl;
  }
}

// ---------------------------------------------------------------------------
// small helper kernels
// ---------------------------------------------------------------------------
__global__ void fill_ones_kernel(float* __restrict__ p, int n) {
  int i = blockIdx.x * blockDim.x + threadIdx.x;
  if (i < n) p[i] = 1.0f;
}

__global__ void count_deg_kernel(float* __restrict__ deg,
                                 const long long* __restrict__ col, int nE) {
  int e = blockIdx.x * blockDim.x + threadIdx.x;
  if (e < nE) unsafeAtomicAdd(&deg[(int)col[e]], 1.0f);
}

__global__ void rsqrt_kernel(float* __restrict__ p, int n) {
  int i = blockIdx.x * blockDim.x + threadIdx.x;
  if (i < n) p[i] = rsqrtf(p[i]);
}

__global__ void zero_kernel(float4* __restrict__ p, int n4) {
  int i = blockIdx.x * blockDim.x + threadIdx.x;
  if (i < n4) p[i] = float4{0.f, 0.f, 0.f, 0.f};
}

// one wave per edge: s[col] += g[row]  (128 floats, float4 per lane)
__global__ __launch_bounds__(256) void scatter_edges_kernel(
    const float* __restrict__ g, float* __restrict__ s,
    const long long* __restrict__ row, const long long* __restrict__ col,
    int nE) {
  int e = blockIdx.x * 8 + (threadIdx.x >> 5);
  if (e >= nE) return;
  int lane = threadIdx.x & 31;
  int r = (int)row[e];  // same address across wave -> HW broadcast
  int c = (int)col[e];
  float4 v = ((const float4*)(g + (size_t)r * NHID))[lane];
  float* sp = s + (size_t)c * NHID + lane * 4;
  unsafeAtomicAdd(sp + 0, v.x);
  unsafeAtomicAdd(sp + 1, v.y);
  unsafeAtomicAdd(sp + 2, v.z);
  unsafeAtomicAdd(sp + 3, v.w);
}

// out = [relu]( dinv[i]*(s + g) + b )
__global__ __launch_bounds__(256) void combine_kernel(
    const float* __restrict__ s, const float* __restrict__ g,
    const float* __restrict__ dinv, const float* __restrict__ b,
    float* __restrict__ out, int M, int do_relu) {
  int idx = blockIdx.x * blockDim.x + threadIdx.x;  // one float4
  int total = M * (NHID / 4);
  if (idx >= total) return;
  int node = idx >> 5;
  int f4 = idx & 31;
  float d = dinv[node];
  float4 sv = ((const float4*)s)[idx];
  float4 gv = ((const float4*)g)[idx];
  float4 bv = ((const float4*)b)[f4];
  float4 o;
  o.x = d * (sv.x + gv.x) + bv.x;
  o.y = d * (sv.y + gv.y) + bv.y;
  o.z = d * (sv.z + gv.z) + bv.z;
  o.w = d * (sv.w + gv.w) + bv.w;
  if (do_relu) {
    o.x = fmaxf(o.x, 0.f);
    o.y = fmaxf(o.y, 0.f);
    o.z = fmaxf(o.z, 0.f);
    o.w = fmaxf(o.w, 0.f);
  }
  ((float4*)out)[idx] = o;
}

// ---------------------------------------------------------------------------
extern "C" void kernel_launch(void* const* d_in, const int* in_sizes, int n_in,
                              void* d_out, int out_size, void* d_ws,
                              size_t ws_size, hipStream_t stream) {
  const float* x = (const float*)d_in[0];        // [M,256]
  const float* W1 = (const float*)d_in[1];       // [256,128]
  const float* b1 = (const float*)d_in[2];       // [128]
  const float* W2 = (const float*)d_in[3];       // [128,128]
  const float* b2 = (const float*)d_in[4];       // [128]
  const long long* ei = (const long long*)d_in[5];  // [2,E] int64

  const int K1 = 256;
  const int M = in_sizes[0] / K1;       // 100000
  const int E = in_sizes[5] / 2;        // 1600000
  const long long* row = ei;
  const long long* col = ei + E;

  // workspace layout
  float* dinv = (float*)d_ws;                       // M floats (padded)
  float* s = dinv + ((M + 1023) & ~1023);           // M*128
  float* g = s + (size_t)M * NHID;                  // M*128
  float* out1 = (float*)d_out;                      // layer-1 activations live in d_out

  const int T = 256;
  // degree -> dinv
  fill_ones_kernel<<<(M + T - 1) / T, T, 0, stream>>>(dinv, M);
  count_deg_kernel<<<(E + T - 1) / T, T, 0, stream>>>(dinv, col, E);
  rsqrt_kernel<<<(M + T - 1) / T, T, 0, stream>>>(dinv, M);

  const int tiles = (M + 15) / 16;
  const int gblocks = (tiles + GEMM_WAVES - 1) / GEMM_WAVES;
  const int n4 = M * (NHID / 4);

  // ---- layer 1 ----
  zero_kernel<<<(n4 + T - 1) / T, T, 0, stream>>>((float4*)s, n4);
  gemm_scale_kernel<<<gblocks, GEMM_WAVES * 32, K1 * NHID * sizeof(float),
                      stream>>>(x, W1, dinv, g, M, K1);
  scatter_edges_kernel<<<(E + 7) / 8, 256, 0, stream>>>(g, s, row, col, E);
  combine_kernel<<<(n4 + T - 1) / T, T, 0, stream>>>(s, g, dinv, b1, out1, M, 1);

  // ---- layer 2 ----
  const int K2 = NHID;
  zero_kernel<<<(n4 + T - 1) / T, T, 0, stream>>>((float4*)s, n4);
  gemm_scale_kernel<<<gblocks, GEMM_WAVES * 32, K2 * NHID * sizeof(float),
                      stream>>>(out1, W2, dinv, g, M, K2);
  scatter_edges_kernel<<<(E + 7) / 8, 256, 0, stream>>>(g, s, row, col, E);
  combine_kernel<<<(n4 + T - 1) / T, T, 0, stream>>>(s, g, dinv, b2,
                                                     (float*)d_out, M, 0);
}